// IntSoftmax_IBERT_30820685316526
// MI455X (gfx1250) — compile-verified
//
#include <hip/hip_runtime.h>
#include <cstdint>

// ---------------------------------------------------------------------------
// I-BERT IntSoftmax, single pass.
//
// Algebraic reduction: the per-tensor QuantAct max is exactly c_int * 2^30
// for any input (row-max element has q=0, r=0; z <= c_int elsewhere), so
// act_scale is computed in closed form from the scaling factor and the whole
// softmax collapses to one streaming pass:
//   201 MB read + 201 MB write = 402 MB  ->  ~17 us at 23.3 TB/s HBM.
//
// One block (256 threads = 8 wave32) per 1024-element row; the row lives in
// registers (float4/thread). Row ingress uses the gfx1250 async global->LDS
// path (ASYNCcnt). q = floor(xi/x0i) is computed by reciprocal-multiply with
// an exact +-1 fixup (bit-identical to IEEE-div + floor for these integer
// operands), removing 4 of the 12 full fp32 division sequences per thread.
// ---------------------------------------------------------------------------

#define AS1 __attribute__((address_space(1)))
#define AS3 __attribute__((address_space(3)))

typedef int v4i __attribute__((ext_vector_type(4)));

#if defined(__HIP_DEVICE_COMPILE__) && defined(__gfx1250__) && \
    __has_builtin(__builtin_amdgcn_global_load_async_to_lds_b128)
#define USE_ASYNC_LDS 1
#else
#define USE_ASYNC_LDS 0
#endif

namespace {
constexpr int   ROW    = 1024;
constexpr int   BLOCK  = 256;              // 8 wave32
constexpr int   WAVES  = BLOCK / 32;
constexpr float X0F    = -0.6931f;                               // -ln(2)
constexpr float C1F    = (float)(0.96963238 / 0.35815147);       // coef1/coef0
constexpr float C2F    = (float)(1.0 / 0.35815147);              // coef2/coef0
constexpr float QMAX   = 32767.0f;                               // 2^15 - 1
constexpr float INV2P25 = 1.0f / 33554432.0f;                    // 2^-25 (exact)
constexpr float OUT_SCALE = 0.0078125f;                          // 2 / 2^8
}

__device__ __forceinline__ float wave_max(float v) {
#pragma unroll
  for (int o = 16; o > 0; o >>= 1) v = fmaxf(v, __shfl_xor(v, o, 32));
  return v;
}
__device__ __forceinline__ float wave_sum(float v) {
#pragma unroll
  for (int o = 16; o > 0; o >>= 1) v += __shfl_xor(v, o, 32);
  return v;
}

__global__ __launch_bounds__(BLOCK) void intsoftmax_ibert_kernel(
    const float* __restrict__ x, const float* __restrict__ sfp,
    float* __restrict__ out, long long n_elems) {
  __shared__ float lds_stage[ROW];
  __shared__ float red_max[WAVES];
  __shared__ float red_sum[WAVES];

  const int  tid  = threadIdx.x;
  const int  lane = tid & 31;
  const int  wave = tid >> 5;
  const long long row = blockIdx.x;

  const float* rowp = x + row * (long long)ROW;
  float*       orow = out + row * (long long)ROW;

  const float sf = *sfp;  // uniform scalar -> s_load

  // ---- row ingress: 16 B per lane ---------------------------------------
  float4 v;
#if USE_ASYNC_LDS
  {
    // CDNA5 async global -> LDS (tracked by ASYNCcnt), then read back our
    // own 16 B. Per-wave wait is sufficient: each lane consumes only the
    // bytes it staged.
    AS1 v4i* g = (AS1 v4i*)(rowp + tid * 4);
    AS3 v4i* l = (AS3 v4i*)(lds_stage + tid * 4);
    __builtin_amdgcn_global_load_async_to_lds_b128(g, l, 0, 0);
#if __has_builtin(__builtin_amdgcn_s_wait_asynccnt)
    __builtin_amdgcn_s_wait_asynccnt(0);
#else
    asm volatile("s_wait_asynccnt 0" ::: "memory");
#endif
    asm volatile("" ::: "memory");
    v = *reinterpret_cast<const float4*>(lds_stage + tid * 4);
  }
#else
  (void)lds_stage;
  v = *reinterpret_cast<const float4*>(rowp + tid * 4);
#endif

  // ---- x_int = trunc(x / sf); row max -----------------------------------
  float xi0 = truncf(v.x / sf);
  float xi1 = truncf(v.y / sf);
  float xi2 = truncf(v.z / sf);
  float xi3 = truncf(v.w / sf);

  float tmax = fmaxf(fmaxf(xi0, xi1), fmaxf(xi2, xi3));
  tmax = wave_max(tmax);
  if (lane == 0) red_max[wave] = tmax;
  __syncthreads();
  float rowmax = red_max[0];
#pragma unroll
  for (int i = 1; i < WAVES; ++i) rowmax = fmaxf(rowmax, red_max[i]);

  // ---- integer exp + closed-form QuantAct(16) ---------------------------
  const float x0i = floorf(X0F / sf);             // negative integer
  const float bi  = floorf(C1F / sf);
  const float ci  = floorf(C2F / (sf * sf));
  // Global max of exp_int is exactly ci * 2^30 (exact fp32: int * pow2).
  const float act_scale = ldexpf(ci, 30) / QMAX;
  const float clampv    = 30.0f * x0i;
  const float inv_x0    = 1.0f / x0i;             // one uniform division

  auto exp_q = [&](float xi) -> float {
    xi -= rowmax;
    xi = fmaxf(xi, clampv);
    // q = floor(xi / x0i) via reciprocal multiply + exact fixup.
    // xi, x0i are integer-valued floats; true residual r = xi - x0i*q lies
    // in (x0i, 0], and fma of these small integers is exact, so the sign
    // tests below recover the exact floor (bit-identical to IEEE div+floor).
    float q = floorf(xi * inv_x0);
    float r = fmaf(-x0i, q, xi);
    if (r > 0.0f)  { q -= 1.0f; r += x0i; }       // overshoot by 1
    if (r <= x0i)  { q += 1.0f; r -= x0i; }       // undershoot by 1
    float z = r * (r + bi) + ci;
    float e = fmaxf(floorf(ldexpf(z, 30 - (int)q)), 0.0f);
    float qv = rintf(e / act_scale);              // round half-to-even
    return fminf(fmaxf(qv, -QMAX), QMAX);
  };

  float e0 = exp_q(xi0);
  float e1 = exp_q(xi1);
  float e2 = exp_q(xi2);
  float e3 = exp_q(xi3);

  // ---- integer normalization --------------------------------------------
  float tsum = (e0 + e1) + (e2 + e3);
  tsum = wave_sum(tsum);
  if (lane == 0) red_sum[wave] = tsum;
  __syncthreads();
  float rowsum = red_sum[0];
#pragma unroll
  for (int i = 1; i < WAVES; ++i) rowsum += red_sum[i];

  const float factor = floorf(4294967296.0f / rowsum);  // floor(2^32 / sum)

  float4 o;
  o.x = floorf(e0 * factor * INV2P25) * OUT_SCALE;
  o.y = floorf(e1 * factor * INV2P25) * OUT_SCALE;
  o.z = floorf(e2 * factor * INV2P25) * OUT_SCALE;
  o.w = floorf(e3 * factor * INV2P25) * OUT_SCALE;
  *reinterpret_cast<float4*>(orow + tid * 4) = o;

  // Trailing tuple element: out_scale scalar.
  if (row == 0 && tid == 0) out[n_elems] = OUT_SCALE;
}

extern "C" void kernel_launch(void* const* d_in, const int* in_sizes, int n_in,
                              void* d_out, int out_size, void* d_ws, size_t ws_size,
                              hipStream_t stream) {
  (void)n_in; (void)out_size; (void)d_ws; (void)ws_size;
  const float* x  = (const float*)d_in[0];
  const float* sf = (const float*)d_in[1];
  float*       out = (float*)d_out;

  const long long n    = (long long)in_sizes[0];   // 4*12*1024*1024
  const int       rows = (int)(n / ROW);           // 49152

  intsoftmax_ibert_kernel<<<rows, BLOCK, 0, stream>>>(x, sf, out, n);
}